// Aggregation_Attn_49443663511676
// MI455X (gfx1250) — compile-verified
//
#include <hip/hip_runtime.h>
#include <hip/hip_bf16.h>
#include <math.h>

// Aggregation attention:  s[n,l] = sum_{t,c} Q[n,t,c,l]*k[t,n,c] ; p = softmax_l(s)
//                         out[t,n,c] = sum_l Q[n,t,c,l]*p[n,l]
// Bandwidth-bound (AI ~ 0.5 FLOP/B): stream Q (128MB) exactly twice; fp32 WMMA
// (V_WMMA_F32_16X16X4_F32) used for the per-n GEMVs to keep exact f32 math while
// hitting the CDNA5 matrix pipe (compute is nowhere near the bottleneck).
//
// WMMA operand layout assumptions (wave32, 32-bit operands):
//   A (16x4, MxK): VGPR0 = {lanes0-15: K=0, lanes16-31: K=2}, VGPR1 = {K=1, K=3},
//                  M = lane&15  (documented in CDNA5 ISA)
//   B (4x16, KxN): assumed symmetric to A: VGPR0 = {K=0 rows lanes0-15, K=2 lanes16-31},
//                  VGPR1 = {K=1, K=3}, N = lane&15  (NOT hardware-verified; we only
//                  broadcast over N, so only the K->VGPR/half-lane mapping matters)
//   C/D (16x16):   VGPR v = {lanes0-15: M=v, N=lane}, {lanes16-31: M=v+8, N=lane-16}

typedef __attribute__((ext_vector_type(2))) float v2f;
typedef __attribute__((ext_vector_type(8))) float v8f;

namespace {
constexpr int N = 32, T = 64, C = 256, L = 64;
constexpr int K = T * C;                 // 16384 reduction rows per n
// pass 1 decomposition
constexpr int SLABS = 64;                // K-slabs per n  -> 32*64 = 2048 blocks
constexpr int KPS   = K / SLABS;         // 256 rows / slab
constexpr int WAVES1 = 8;
constexpr int KPW   = KPS / WAVES1;      // 32 rows / wave
// pass 2 decomposition
constexpr int ROWS_PER_WAVE  = 16;
constexpr int WAVES2         = 8;
constexpr int ROWS_PER_BLOCK = ROWS_PER_WAVE * WAVES2;  // 128
constexpr int BLOCKS_PER_N   = K / ROWS_PER_BLOCK;      // 128 -> 4096 blocks
} // namespace

// ---------------- Pass 1: partial scores  ws_partial[n][slab][64] ----------------
__global__ __launch_bounds__(256) void aggattn_scores(const float* __restrict__ q,
                                                      const float* __restrict__ key,
                                                      float* __restrict__ partial) {
  __shared__ float lds[WAVES1][L];
  const int nb   = blockIdx.x / SLABS;
  const int slab = blockIdx.x % SLABS;
  const int tid  = threadIdx.x;
  const int w    = tid >> 5;
  const int lane = tid & 31;
  const int m    = lane & 15;     // A-matrix row (= l within tile)
  const int a    = lane >> 4;     // half-wave selects K pair {0,1} vs {2,3}

  const float* qn = q + (size_t)nb * K * L;
  const int kw = slab * KPS + w * KPW;

  v8f acc[4] = {v8f{}, v8f{}, v8f{}, v8f{}};

  for (int i = 0; i < KPW; i += 4) {
    const int k0 = kw + i;
    // key[t,n,c], k = t*C + c  -> 4 consecutive c within one t (k0 % 4 == 0)
    const float* kp = key + (size_t)(k0 >> 8) * (N * C) + nb * C + (k0 & 255);
    const float kv0 = kp[0], kv1 = kp[1], kv2 = kp[2], kv3 = kp[3];
    v2f B;                               // broadcast key over all 16 N columns
    B.x = a ? kv2 : kv0;                 // K=2 : K=0
    B.y = a ? kv3 : kv1;                 // K=3 : K=1

    // A[m][kk] = Q[(k0+kk)*L + l0 + m]; this lane supplies kk = 2a, 2a+1
    const float* qa = qn + (size_t)(k0 + 2 * a) * L + m;
    __builtin_prefetch(qa + 4 * L, 0, 1);  // next K-chunk -> global_prefetch_b8
#pragma unroll
    for (int lt = 0; lt < 4; ++lt) {     // l tiles: l0 = 16*lt
      v2f A;
      A.x = qa[lt * 16];                 // K = 2a
      A.y = qa[lt * 16 + L];             // K = 2a+1
      acc[lt] = __builtin_amdgcn_wmma_f32_16x16x4_f32(
          false, A, false, B, (short)0, acc[lt], false, false);
    }
  }

  // every D column is identical (B columns identical): read column 0 (lanes 0,16)
  if (m == 0) {
#pragma unroll
    for (int lt = 0; lt < 4; ++lt)
#pragma unroll
      for (int v = 0; v < 8; ++v)
        lds[w][lt * 16 + a * 8 + v] = acc[lt][v];
  }
  __syncthreads();

  if (tid < L) {
    float s = 0.f;
#pragma unroll
    for (int ww = 0; ww < WAVES1; ++ww) s += lds[ww][tid];
    partial[((size_t)nb * SLABS + slab) * L + tid] = s;
  }
}

// ---------------- Pass 1b: reduce slabs + softmax -> probs[n][64] ----------------
__global__ __launch_bounds__(64) void aggattn_softmax(const float* __restrict__ partial,
                                                      float* __restrict__ probs) {
  __shared__ float s_sc[L];
  __shared__ float s_e[L];
  const int nb = blockIdx.x;
  const int tid = threadIdx.x;

  float s = 0.f;
  for (int sl = 0; sl < SLABS; ++sl)
    s += partial[((size_t)nb * SLABS + sl) * L + tid];
  s_sc[tid] = s;
  __syncthreads();

  float mx = -INFINITY;                  // fixed-order scan: deterministic
#pragma unroll
  for (int i = 0; i < L; ++i) mx = fmaxf(mx, s_sc[i]);
  const float e = expf(s - mx);
  s_e[tid] = e;
  __syncthreads();

  float sum = 0.f;
#pragma unroll
  for (int i = 0; i < L; ++i) sum += s_e[i];
  probs[(size_t)nb * L + tid] = e / sum;
}

// ---------------- Pass 2: out[t,n,c] = sum_l Q[n,t,c,l] * p[n,l] ----------------
__global__ __launch_bounds__(256) void aggattn_out(const float* __restrict__ q,
                                                   const float* __restrict__ probs,
                                                   float* __restrict__ out) {
  __shared__ float p[L];
  const int nb = blockIdx.x / BLOCKS_PER_N;
  const int rb = blockIdx.x % BLOCKS_PER_N;
  const int tid  = threadIdx.x;
  const int w    = tid >> 5;
  const int lane = tid & 31;
  const int m    = lane & 15;            // A row (= output row within 16-row tile)
  const int a    = lane >> 4;

  if (tid < L) p[tid] = probs[(size_t)nb * L + tid];
  __syncthreads();

  const int k0 = rb * ROWS_PER_BLOCK + w * ROWS_PER_WAVE;
  const float* qn = q + (size_t)nb * K * L;
  // A[m][kk] = Q[(k0+m)*L + l0 + kk]; lane supplies kk = 2a, 2a+1 (contiguous pair)
  const float* qa = qn + (size_t)(k0 + m) * L + 2 * a;

  v8f acc = v8f{};
#pragma unroll
  for (int l0 = 0; l0 < L; l0 += 4) {
    const v2f A = *(const v2f*)(qa + l0);  // 8B-aligned pair -> global_load_b64
    v2f B;                                 // broadcast p over all 16 N columns
    B.x = a ? p[l0 + 2] : p[l0 + 0];
    B.y = a ? p[l0 + 3] : p[l0 + 1];
    acc = __builtin_amdgcn_wmma_f32_16x16x4_f32(
        false, A, false, B, (short)0, acc, false, false);
  }

  // D column 0 carries the 16 output rows; rows k0..k0+15 share t, consecutive c
  if (m == 0) {
    float* ob = out + (size_t)(k0 >> 8) * (N * C) + nb * C + (k0 & 255) + a * 8;
#pragma unroll
    for (int v = 0; v < 8; ++v) ob[v] = acc[v];
  }
}

extern "C" void kernel_launch(void* const* d_in, const int* in_sizes, int n_in,
                              void* d_out, int out_size, void* d_ws, size_t ws_size,
                              hipStream_t stream) {
  const float* q   = (const float*)d_in[0];   // (32, 64, 256, 64) f32
  const float* key = (const float*)d_in[1];   // (64, 32, 256)     f32
  float* out = (float*)d_out;                 // (64, 32, 256)     f32

  float* partial = (float*)d_ws;              // [32][64][64]  = 512 KB
  float* probs   = partial + (size_t)N * SLABS * L;  // [32][64] = 8 KB

  aggattn_scores<<<dim3(N * SLABS), dim3(256), 0, stream>>>(q, key, partial);
  aggattn_softmax<<<dim3(N), dim3(64), 0, stream>>>(partial, probs);
  aggattn_out<<<dim3(N * BLOCKS_PER_N), dim3(256), 0, stream>>>(q, probs, out);
}